// Net_59511066853411
// MI455X (gfx1250) — compile-verified
//
#include <hip/hip_runtime.h>
#include <math.h>

#define F    1433
#define FP   1440      // F padded to multiple of 16 (and 8 for K-unroll)
#define BB   64        // batch
#define K    8         // neighbors
#define HID  2
#define NC   7
#define EPS  1e-5f

typedef float v2f __attribute__((ext_vector_type(2)));
typedef float v8f __attribute__((ext_vector_type(8)));

// ---------------------------------------------------------------------------
// 0) ys[b,f] = sum_k neighbor[b,k,f]; also zero the 8 stats scalars.
__global__ void prep_kernel(const float* __restrict__ nb, float* __restrict__ ys,
                            float* __restrict__ scal) {
  int i = blockIdx.x * blockDim.x + threadIdx.x;
  if (i < 8) scal[i] = 0.f;
  if (i >= BB * F) return;
  int b = i / F, f = i % F;
  float s = 0.f;
  #pragma unroll
  for (int k = 0; k < K; ++k) s += nb[(b * K + k) * F + f];
  ys[i] = s;
}

// ---------------------------------------------------------------------------
// 1) invc[b,g] = 1 / sum_f sqrt(xs[b,f]*ys[b,g] + xs[b,g]*ys[b,f])   (0 if sum==0)
//    One block per (8 columns g, batch b); rows staged once in LDS.
__global__ void __launch_bounds__(256)
colsum_kernel(const float* __restrict__ xs, const float* __restrict__ ys,
              float* __restrict__ invc) {
  __shared__ float sx[FP], sy[FP];
  int b = blockIdx.y;
  for (int i = threadIdx.x; i < FP; i += 256) {
    sx[i] = (i < F) ? xs[b * F + i] : 0.f;
    sy[i] = (i < F) ? ys[b * F + i] : 0.f;
  }
  __syncthreads();
  int wv = threadIdx.x >> 5, lane = threadIdx.x & 31;
  int g = blockIdx.x * 8 + wv;
  if (g >= F) return;
  float xg = sx[g], yg = sy[g];
  float s = 0.f;
  for (int f = lane; f < F; f += 32)
    s += sqrtf(sx[f] * yg + xg * sy[f]);
  for (int off = 16; off; off >>= 1) s += __shfl_down(s, off, 32);
  if (lane == 0) invc[b * F + g] = (s > 0.f) ? 1.f / s : 0.f;
}

// ---------------------------------------------------------------------------
// 2) y1[b,f] = sum_g sqrt(xf*yg + xg*yf) * (xg * invc[b,g]) + xf
//    Rows (and xs*invc weight row) staged in LDS; BN1 stats via block atomics.
__global__ void __launch_bounds__(256)
y1_kernel(const float* __restrict__ xs, const float* __restrict__ ys,
          const float* __restrict__ invc, float* __restrict__ y1,
          float* __restrict__ scal) {
  __shared__ float sx[FP], sy[FP], sw[FP];
  __shared__ float smY[8], smY2[8];
  int b = blockIdx.y;
  for (int i = threadIdx.x; i < FP; i += 256) {
    float xv = (i < F) ? xs[b * F + i] : 0.f;
    sx[i] = xv;
    sy[i] = (i < F) ? ys[b * F + i] : 0.f;
    sw[i] = (i < F) ? xv * invc[b * F + i] : 0.f;   // xg / colsum[b,g]
  }
  __syncthreads();
  int wv = threadIdx.x >> 5, lane = threadIdx.x & 31;
  int fidx = blockIdx.x * 8 + wv;
  float y = 0.f;
  if (fidx < F) {
    float xf = sx[fidx], yf = sy[fidx];
    float s = 0.f;
    for (int g = lane; g < F; g += 32)
      s += sqrtf(xf * sy[g] + sx[g] * yf) * sw[g];
    for (int off = 16; off; off >>= 1) s += __shfl_down(s, off, 32);
    y = s + xf;
    if (lane == 0) y1[b * F + fidx] = y;
  }
  if (lane == 0) { smY[wv] = y; smY2[wv] = y * y; }   // y==0 when fidx>=F
  __syncthreads();
  if (threadIdx.x == 0) {
    float a = 0.f, a2 = 0.f;
    #pragma unroll
    for (int w = 0; w < 8; ++w) { a += smY[w]; a2 += smY2[w]; }
    atomicAdd(&scal[0], a);
    atomicAdd(&scal[1], a2);
  }
}

// ---------------------------------------------------------------------------
// 3) adjT[f,g] = sum_b sqrt(xs[b,f]*ys[b,g] + xs[b,g]*ys[b,f]) * invc[b,g]
//               + adj_buf[f,g] + BB*(f==g),   zero-padded to FP x FP.
//    16x16 (f,g) tile per 256-thread block; operands staged in LDS.
__global__ void __launch_bounds__(256)
adjtot_kernel(const float* __restrict__ xs, const float* __restrict__ ys,
              const float* __restrict__ invc, const float* __restrict__ adj_buf,
              float* __restrict__ adjT) {
  __shared__ float sxf[BB * 16], syf[BB * 16], sxg[BB * 16], syg[BB * 16], sic[BB * 16];
  int f0 = blockIdx.x * 16, g0 = blockIdx.y * 16;
  int t = threadIdx.x;
  #pragma unroll
  for (int i = 0; i < 4; ++i) {
    int e = t + i * 256;      // [0,1024)
    int b = e >> 4, j = e & 15;
    int fi = f0 + j, gi = g0 + j;
    sxf[e] = (fi < F) ? xs[b * F + fi] : 0.f;
    syf[e] = (fi < F) ? ys[b * F + fi] : 0.f;
    sxg[e] = (gi < F) ? xs[b * F + gi] : 0.f;
    syg[e] = (gi < F) ? ys[b * F + gi] : 0.f;
    sic[e] = (gi < F) ? invc[b * F + gi] : 0.f;
  }
  __syncthreads();
  int tf = t & 15, tg = t >> 4;
  int f = f0 + tf, g = g0 + tg;
  float acc = 0.f;
  if (f < F && g < F) {
    for (int b = 0; b < BB; ++b)
      acc += sqrtf(sxf[b * 16 + tf] * syg[b * 16 + tg] +
                   sxg[b * 16 + tg] * syf[b * 16 + tf]) * sic[b * 16 + tg];
    acc += adj_buf[f * F + g];
    if (f == g) acc += (float)BB;
  }
  adjT[f * FP + g] = acc;
}

// ---------------------------------------------------------------------------
// 4) BN1 + softsign into padded GEMM B-matrix: Bm[g, n], n = 2*b + o, g < FP.
//    h1 = W1[o]*y1 + b1[o] is affine -> stats derive from (sumY, sumY2).
__global__ void bn1_softsign_kernel(const float* __restrict__ y1, const float* __restrict__ W1,
                                    const float* __restrict__ g1, const float* __restrict__ be1,
                                    const float* __restrict__ scal, float* __restrict__ Bm) {
  int idx = blockIdx.x * blockDim.x + threadIdx.x;   // [0, FP*128)
  if (idx >= FP * 128) return;
  int g = idx >> 7, n = idx & 127;
  float out = 0.f;
  if (g < F) {
    int b = n >> 1, o = n & 1;
    const float N = (float)(BB * F);
    float muY  = scal[0] / N;
    float varY = scal[1] / N - muY * muY;
    float w  = W1[o];
    float rs = rsqrtf(w * w * varY + EPS);           // var(h1) = W1^2 * var(y1)
    float hn = (w * (y1[b * F + g] - muY)) * rs * g1[o] + be1[o];
    out = hn / (1.f + fabsf(hn));
  }
  Bm[idx] = out;
}

// ---------------------------------------------------------------------------
// 5) GEMM via V_WMMA_F32_16X16X4_F32:  D[f,n] = sum_g adjT[f,g] * Bm[g,n]
//    M=FP, N=128, K=FP. One 256-thread block per f-tile; 8 waves = 8 n-tiles.
//    A panel (16x32, stride 33 to avoid bank conflicts) staged as one aligned
//    float2 per thread; B panel (32x128) staged as bulk float4 copies.
//    A frag (16x4 f32): lane l<16 holds A[M=l][K=0],[K=1]; lane l+16: K=2,3.
//    B frag (4x16 f32): lane l<16 holds B[K=0][N=l],B[K=1][N=l]; l+16: K=2,3.
//    C/D: VGPR r, lanes 0-15 -> (M=r, N=l); lanes 16-31 -> (M=r+8, N=l).
#define ASTR 33
__global__ void __launch_bounds__(256)
wmma_gemm_kernel(const float* __restrict__ adjT, const float* __restrict__ Bm,
                 float* __restrict__ D) {
  __shared__ float sA[16 * ASTR];                   // [f 0..15][k 0..31], padded stride
  __shared__ __align__(16) float sB[32 * 128];      // [k 0..31][n 0..127]
  int f0 = blockIdx.x * 16;
  int wv = threadIdx.x >> 5;          // n-tile index 0..7
  int lane = threadIdx.x & 31;
  int half = lane >> 4;
  int l    = lane & 15;
  int n0   = wv * 16;
  // Per-thread fixed staging coordinates for the A panel (2 floats each).
  int afr = threadIdx.x >> 4;         // row 0..15
  int akk = (threadIdx.x & 15) * 2;   // k-pair 0,2,...,30 (8B aligned in row)
  const float4* bsrc4 = (const float4*)Bm;   // Bm rows are 16B aligned (128 floats)
  float4* sB4 = (float4*)sB;
  v8f c0 = {}; v8f c1 = {};
  for (int k0 = 0; k0 < FP; k0 += 32) {
    // stage A: one aligned float2 per thread
    v2f av = *(const v2f*)&adjT[(size_t)(f0 + afr) * FP + (k0 + akk)];
    sA[afr * ASTR + akk]     = av.x;
    sA[afr * ASTR + akk + 1] = av.y;
    // prefetch next A panel (speculative; dropped past the end)
    __builtin_prefetch(&adjT[(size_t)(f0 + afr) * FP + (k0 + akk) + 32], 0, 0);
    // stage B: 4096 floats = 1024 float4, 4 per thread (coalesced b128)
    #pragma unroll
    for (int i = 0; i < 4; ++i)
      sB4[threadIdx.x + i * 256] = bsrc4[(size_t)(k0 >> 2) * 128 + threadIdx.x + i * 256];
    __syncthreads();
    #pragma unroll
    for (int kk = 0; kk < 32; kk += 8) {
      int ka = kk + half * 2;
      v2f a0; a0.x = sA[l * ASTR + ka];       a0.y = sA[l * ASTR + ka + 1];
      v2f b0; b0.x = sB[ka * 128 + n0 + l];   b0.y = sB[(ka + 1) * 128 + n0 + l];
      c0 = __builtin_amdgcn_wmma_f32_16x16x4_f32(false, a0, false, b0, (short)0, c0, false, false);
      int kb = kk + 4 + half * 2;
      v2f a1; a1.x = sA[l * ASTR + kb];       a1.y = sA[l * ASTR + kb + 1];
      v2f b1; b1.x = sB[kb * 128 + n0 + l];   b1.y = sB[(kb + 1) * 128 + n0 + l];
      c1 = __builtin_amdgcn_wmma_f32_16x16x4_f32(false, a1, false, b1, (short)0, c1, false, false);
    }
    __syncthreads();
  }
  #pragma unroll
  for (int r = 0; r < 8; ++r)
    D[(size_t)(f0 + r + 8 * half) * 128 + (n0 + l)] = c0[r] + c1[r];
}

// ---------------------------------------------------------------------------
// 6) BN2 batch statistics per channel: h2[b,o,f] = W2[o,0]*y2_0 + W2[o,1]*y2_1 + b2[o]
__global__ void bn2_stats_kernel(const float* __restrict__ D, const float* __restrict__ W2,
                                 const float* __restrict__ b2, float* __restrict__ scal) {
  __shared__ float sm[8][4];
  float s0 = 0.f, s00 = 0.f, s1 = 0.f, s11 = 0.f;
  for (int i = blockIdx.x * blockDim.x + threadIdx.x; i < BB * F;
       i += gridDim.x * blockDim.x) {
    int b = i / F, f = i % F;
    float y0 = D[f * 128 + 2 * b], y1v = D[f * 128 + 2 * b + 1];
    float h0 = W2[0] * y0 + W2[1] * y1v + b2[0];
    float h1 = W2[2] * y0 + W2[3] * y1v + b2[1];
    s0 += h0; s00 += h0 * h0; s1 += h1; s11 += h1 * h1;
  }
  for (int off = 16; off; off >>= 1) {
    s0  += __shfl_down(s0,  off, 32);
    s00 += __shfl_down(s00, off, 32);
    s1  += __shfl_down(s1,  off, 32);
    s11 += __shfl_down(s11, off, 32);
  }
  int lane = threadIdx.x & 31, wv = threadIdx.x >> 5;
  if (lane == 0) { sm[wv][0] = s0; sm[wv][1] = s00; sm[wv][2] = s1; sm[wv][3] = s11; }
  __syncthreads();
  if (threadIdx.x < 4) {
    float acc = 0.f;
    #pragma unroll
    for (int w = 0; w < 8; ++w) acc += sm[w][threadIdx.x];
    atomicAdd(&scal[2 + threadIdx.x], acc);
  }
}

// ---------------------------------------------------------------------------
// 7) Fused BN2 + softsign + classifier: out[b,c] = sum_{o,f} s2[b,o,f]*Wc[c,o*F+f] + bc[c]
__global__ void classify_kernel(const float* __restrict__ D, const float* __restrict__ W2,
                                const float* __restrict__ b2, const float* __restrict__ g2,
                                const float* __restrict__ be2, const float* __restrict__ Wc,
                                const float* __restrict__ bc, const float* __restrict__ scal,
                                float* __restrict__ out) {
  __shared__ float sm[8][NC];
  int b = blockIdx.x;
  const float N = (float)(BB * F);
  float mu0 = scal[2] / N, var0 = scal[3] / N - mu0 * mu0;
  float mu1 = scal[4] / N, var1 = scal[5] / N - mu1 * mu1;
  float rs0 = rsqrtf(var0 + EPS), rs1 = rsqrtf(var1 + EPS);
  float acc[NC];
  #pragma unroll
  for (int c = 0; c < NC; ++c) acc[c] = 0.f;
  for (int f = threadIdx.x; f < F; f += blockDim.x) {
    float y0 = D[f * 128 + 2 * b], y1v = D[f * 128 + 2 * b + 1];
    float h0 = W2[0] * y0 + W2[1] * y1v + b2[0];
    float h1 = W2[2] * y0 + W2[3] * y1v + b2[1];
    float hn0 = (h0 - mu0) * rs0 * g2[0] + be2[0];
    float hn1 = (h1 - mu1) * rs1 * g2[1] + be2[1];
    float s0 = hn0 / (1.f + fabsf(hn0));
    float s1 = hn1 / (1.f + fabsf(hn1));
    #pragma unroll
    for (int c = 0; c < NC; ++c)
      acc[c] += s0 * Wc[c * (2 * F) + f] + s1 * Wc[c * (2 * F) + F + f];
  }
  #pragma unroll
  for (int c = 0; c < NC; ++c)
    for (int off = 16; off; off >>= 1) acc[c] += __shfl_down(acc[c], off, 32);
  int lane = threadIdx.x & 31, wv = threadIdx.x >> 5;
  if (lane == 0)
    #pragma unroll
    for (int c = 0; c < NC; ++c) sm[wv][c] = acc[c];
  __syncthreads();
  if (threadIdx.x < NC) {
    float t = 0.f;
    #pragma unroll
    for (int w = 0; w < 8; ++w) t += sm[w][threadIdx.x];
    out[b * NC + threadIdx.x] = t + bc[threadIdx.x];
  }
}

// ---------------------------------------------------------------------------
extern "C" void kernel_launch(void* const* d_in, const int* in_sizes, int n_in,
                              void* d_out, int out_size, void* d_ws, size_t ws_size,
                              hipStream_t stream) {
  (void)in_sizes; (void)n_in; (void)out_size; (void)ws_size;
  const float* x    = (const float*)d_in[0];   // [B,1,F] == xs [B,F]
  const float* nb   = (const float*)d_in[1];   // [B,K,F]
  const float* abuf = (const float*)d_in[2];   // [1,F,F] (zeros, but honored)
  const float* W1   = (const float*)d_in[3];
  const float* g1   = (const float*)d_in[7];
  const float* be1  = (const float*)d_in[8];
  const float* W2   = (const float*)d_in[5];
  const float* b2   = (const float*)d_in[6];
  const float* g2   = (const float*)d_in[9];
  const float* be2  = (const float*)d_in[10];
  const float* Wc   = (const float*)d_in[11];
  const float* bc   = (const float*)d_in[12];
  float* out = (float*)d_out;

  // workspace layout (floats): total ~2.72M floats ~= 10.9 MB
  float* ws    = (float*)d_ws;
  float* adjT  = ws;                       // FP*FP
  float* Bm    = adjT + (size_t)FP * FP;   // FP*128
  float* Dm    = Bm   + (size_t)FP * 128;  // FP*128
  float* ys    = Dm   + (size_t)FP * 128;  // B*F
  float* invc  = ys   + (size_t)BB * F;    // B*F
  float* y1    = invc + (size_t)BB * F;    // B*F
  float* scal  = y1   + (size_t)BB * F;    // 8 scalars

  const int colBlocks = (F + 7) / 8;       // 180 column-groups of 8 waves

  prep_kernel<<<(BB * F + 255) / 256, 256, 0, stream>>>(nb, ys, scal);
  colsum_kernel<<<dim3(colBlocks, BB), 256, 0, stream>>>(x, ys, invc);
  y1_kernel<<<dim3(colBlocks, BB), 256, 0, stream>>>(x, ys, invc, y1, scal);
  adjtot_kernel<<<dim3(FP / 16, FP / 16), 256, 0, stream>>>(x, ys, invc, abuf, adjT);
  bn1_softsign_kernel<<<(FP * 128) / 256, 256, 0, stream>>>(y1, W1, g1, be1, scal, Bm);
  wmma_gemm_kernel<<<FP / 16, 256, 0, stream>>>(adjT, Bm, Dm);
  bn2_stats_kernel<<<128, 256, 0, stream>>>(Dm, W2, b2, scal);
  classify_kernel<<<BB, 256, 0, stream>>>(Dm, W2, b2, g2, be2, Wc, bc, scal, out);
}